// DictLearn_18159121727816
// MI455X (gfx1250) — compile-verified
//
#include <hip/hip_runtime.h>
#include <hip/hip_bf16.h>
#include <math.h>

// ---------------------------------------------------------------------------
// DictLearn (IHT sparse coding) on MI455X / gfx1250.
// fp32 WMMA (v_wmma_f32_16x16x4_f32) for all GEMMs; data is L2-resident.
// K-loop is explicitly software-pipelined (double-buffered fragments).
// ---------------------------------------------------------------------------

typedef __attribute__((ext_vector_type(2))) float v2f;
typedef __attribute__((ext_vector_type(8))) float v8f;

#define N_DIM 512
#define M_DIM 1024
#define B_DIM 1024
#define IHT_STEPS 50
#define POWER_STEPS 100

__device__ __forceinline__ v8f wmma_f32_k4(v2f a, v2f b, v8f c) {
  // D = A(16x4) * B(4x16) + C(16x16), fp32
  return __builtin_amdgcn_wmma_f32_16x16x4_f32(
      /*neg_a=*/false, a, /*neg_b=*/false, b,
      /*c_mod=*/(short)0, c, /*reuse_a=*/false, /*reuse_b=*/false);
}

// ---------------------------------------------------------------------------
// Column norms of W [512 x 1024]: cn[m] = 1 / ||W[:,m]||
// ---------------------------------------------------------------------------
__global__ void colnorm_kernel(const float* __restrict__ W, float* __restrict__ cn) {
  int m = blockIdx.x * 256 + threadIdx.x;
  if (m >= M_DIM) return;
  float s = 0.0f;
  for (int n = 0; n < N_DIM; ++n) {
    float v = W[(size_t)n * M_DIM + m];
    s += v * v;
  }
  cn[m] = 1.0f / sqrtf(s);
}

// Wn[n][m] = W[n][m]*cn[m];  WnT[m][n] = Wn[n][m]
__global__ void normalize_kernel(const float* __restrict__ W, const float* __restrict__ cn,
                                 float* __restrict__ Wn, float* __restrict__ WnT) {
  int idx = blockIdx.x * 256 + threadIdx.x;
  if (idx >= N_DIM * M_DIM) return;
  int n = idx >> 10;          // / M_DIM
  int m = idx & (M_DIM - 1);  // % M_DIM
  float v = W[idx] * cn[m];
  Wn[idx] = v;
  WnT[(size_t)m * N_DIM + n] = v;
}

// ---------------------------------------------------------------------------
// Power method: x <- (x Wn^T) Wn / ||.||, 100 iters; eta = 1/last_norm.
// Single 1024-thread workgroup (32 waves). Wn is L2-resident (2 MB).
// ---------------------------------------------------------------------------
__global__ void __launch_bounds__(1024) power_method_kernel(const float* __restrict__ Wn,
                                                            float* __restrict__ etaOut) {
  __shared__ float x[M_DIM];
  __shared__ float t[N_DIM];
  __shared__ float red[1024];
  const int tid = threadIdx.x;
  const int wave = tid >> 5;
  const int lane = tid & 31;

  // Deterministic pseudo-random init (power method converges from any generic x0)
  unsigned u = (unsigned)tid * 2654435761u + 12345u;
  u ^= u >> 13; u *= 2246822519u; u ^= u >> 16;
  x[tid] = ((u >> 9) * (1.0f / 8388608.0f)) - 1.0f;
  __syncthreads();

  float nm = 1.0f;
  for (int it = 0; it < POWER_STEPS; ++it) {
    // t[n] = sum_m x[m] * Wn[n][m] : one wave per group of 16 rows, lane-parallel m
    for (int i = 0; i < 16; ++i) {
      int n = wave * 16 + i;
      const float* wr = Wn + (size_t)n * M_DIM;
      float p = 0.0f;
      for (int m = lane; m < M_DIM; m += 32) p += x[m] * wr[m];
      for (int off = 16; off > 0; off >>= 1) p += __shfl_xor(p, off, 32);
      if (lane == 0) t[n] = p;
    }
    __syncthreads();
    // x2[m] = sum_n t[n] * Wn[n][m]   (coalesced column reads)
    float s = 0.0f;
    for (int n = 0; n < N_DIM; ++n) s += t[n] * Wn[(size_t)n * M_DIM + tid];
    red[tid] = s * s;
    __syncthreads();
    for (int off = 512; off > 0; off >>= 1) {
      if (tid < off) red[tid] += red[tid + off];
      __syncthreads();
    }
    nm = sqrtf(red[0]);
    __syncthreads();
    x[tid] = s / nm;
    __syncthreads();
  }
  if (tid == 0) etaOut[0] = 1.0f / nm;
}

// ---------------------------------------------------------------------------
// Deterministic two-stage sum-of-squares reduction
// ---------------------------------------------------------------------------
__global__ void sq_partial_kernel(const float* __restrict__ src, int n,
                                  float* __restrict__ partial) {
  __shared__ float red[256];
  float s = 0.0f;
  for (int i = blockIdx.x * 256 + threadIdx.x; i < n; i += 256 * gridDim.x) {
    float v = src[i];
    s += v * v;
  }
  red[threadIdx.x] = s;
  __syncthreads();
  for (int off = 128; off > 0; off >>= 1) {
    if ((int)threadIdx.x < off) red[threadIdx.x] += red[threadIdx.x + off];
    __syncthreads();
  }
  if (threadIdx.x == 0) partial[blockIdx.x] = red[0];
}

__global__ void sq_finalize_kernel(const float* __restrict__ partial, int nparts,
                                   float* __restrict__ out,
                                   const float* __restrict__ ynSlot, int divide) {
  __shared__ float red[256];
  float s = 0.0f;
  for (int i = threadIdx.x; i < nparts; i += 256) s += partial[i];
  red[threadIdx.x] = s;
  __syncthreads();
  for (int off = 128; off > 0; off >>= 1) {
    if ((int)threadIdx.x < off) red[threadIdx.x] += red[threadIdx.x + off];
    __syncthreads();
  }
  if (threadIdx.x == 0) {
    float r = sqrtf(red[0]);
    if (divide) r /= ynSlot[0];
    out[0] = r;
  }
}

// ---------------------------------------------------------------------------
// WMMA fp32 GEMM: C[m][n] = combine( sum_k A[m*lda+k] * Bt[n*ldbt+k] )
//   mode 0: C = alpha * AB          (init Gamma pre-threshold)
//   mode 1: C = Cin - alpha * AB    (IHT gradient step)
//   mode 2: C = AB - Ysub           (residual)
//   mode 3: C = AB                  (final X)
// Block = 128 threads (4 waves), arranged 2(m) x 2(n).
// Wave tile = 32 x (16*NT): NT=2 -> 4 loads / 4 WMMA; NT=4 -> 6 loads / 8 WMMA.
// K-loop is double-buffered: next k-step's fragments are loaded before the
// current k-step's WMMAs so L2 latency overlaps the matrix pipe.
// Fragment layouts per CDNA5 ISA:
//   A 16x4 f32: lane L<16 -> row L, K={0,1}; lane L>=16 -> row L-16, K={2,3}
//   B 4x16 f32: symmetric (lane = N column, K along vgpr x lane-half)
//   C/D 16x16:  vgpr r, lanes 0-15 -> M=r, N=lane; lanes 16-31 -> M=r+8
// ---------------------------------------------------------------------------
template <int NT>
__global__ void __launch_bounds__(128) gemm_wmma_kernel(
    float* __restrict__ C, int ldc,
    const float* __restrict__ A, int lda,
    const float* __restrict__ Bt, int ldbt,
    const float* __restrict__ Cin,
    const float* __restrict__ Ysub,
    const float* __restrict__ alphaPtr,
    int Kdim, int mode) {
  const int tid  = threadIdx.x;
  const int wave = tid >> 5;
  const int lane = tid & 31;
  const int half = lane >> 4;   // which K-pair half
  const int lo   = lane & 15;   // row (A) / col (B) within tile

  const int m0 = blockIdx.y * 64 + (wave >> 1) * 32;
  const int n0 = blockIdx.x * (32 * NT) + (wave & 1) * (16 * NT);

  const float* aRow[2];
  aRow[0] = A + (size_t)(m0 + lo) * lda + 2 * half;
  aRow[1] = aRow[0] + (size_t)16 * lda;
  const float* bRow[NT];
#pragma unroll
  for (int j = 0; j < NT; ++j)
    bRow[j] = Bt + (size_t)(n0 + 16 * j + lo) * ldbt + 2 * half;

  v8f acc[2][NT];
#pragma unroll
  for (int i = 0; i < 2; ++i)
#pragma unroll
    for (int j = 0; j < NT; ++j) acc[i][j] = {};

  // prologue: fragments for k = 0
  v2f aCur[2], bCur[NT];
#pragma unroll
  for (int i = 0; i < 2; ++i) aCur[i] = *(const v2f*)(aRow[i]);
#pragma unroll
  for (int j = 0; j < NT; ++j) bCur[j] = *(const v2f*)(bRow[j]);

#pragma unroll 2
  for (int k = 4; k < Kdim; k += 4) {
    v2f aNxt[2], bNxt[NT];
#pragma unroll
    for (int i = 0; i < 2; ++i) aNxt[i] = *(const v2f*)(aRow[i] + k);
#pragma unroll
    for (int j = 0; j < NT; ++j) bNxt[j] = *(const v2f*)(bRow[j] + k);
#pragma unroll
    for (int i = 0; i < 2; ++i)
#pragma unroll
      for (int j = 0; j < NT; ++j)
        acc[i][j] = wmma_f32_k4(aCur[i], bCur[j], acc[i][j]);
#pragma unroll
    for (int i = 0; i < 2; ++i) aCur[i] = aNxt[i];
#pragma unroll
    for (int j = 0; j < NT; ++j) bCur[j] = bNxt[j];
  }
  // epilogue k-step
#pragma unroll
  for (int i = 0; i < 2; ++i)
#pragma unroll
    for (int j = 0; j < NT; ++j)
      acc[i][j] = wmma_f32_k4(aCur[i], bCur[j], acc[i][j]);

  const float alpha = alphaPtr[0];

#pragma unroll
  for (int mt = 0; mt < 2; ++mt) {
#pragma unroll
    for (int nt = 0; nt < NT; ++nt) {
      const int rbase = m0 + 16 * mt + 8 * half;
      const int col   = n0 + 16 * nt + lo;
#pragma unroll
      for (int r = 0; r < 8; ++r) {
        size_t off = (size_t)(rbase + r) * ldc + col;
        float v = acc[mt][nt][r];
        float outv;
        if (mode == 0)      outv = alpha * v;
        else if (mode == 1) outv = Cin[off] - alpha * v;
        else if (mode == 2) outv = v - Ysub[off];
        else                outv = v;
        C[off] = outv;
      }
    }
  }
}

// ---------------------------------------------------------------------------
// Per-row hard threshold: keep entries with |g| > (K+1)-th largest |g|.
// One 256-thread block per row; in-LDS bitonic sort of 1024 abs values.
// ---------------------------------------------------------------------------
__global__ void __launch_bounds__(256) hard_threshold_kernel(
    const float* __restrict__ Pre, float* __restrict__ G,
    const int* __restrict__ Kptr) {
  __shared__ float s[M_DIM];
  const int row = blockIdx.x;
  const int tid = threadIdx.x;
  const float* src = Pre + (size_t)row * M_DIM;

  for (int j = tid; j < M_DIM; j += 256) s[j] = fabsf(src[j]);
  __syncthreads();

  // Bitonic sort ascending
  for (int k = 2; k <= M_DIM; k <<= 1) {
    for (int j = k >> 1; j > 0; j >>= 1) {
      for (int base = 0; base < M_DIM; base += 256) {
        int i = base + tid;
        int ixj = i ^ j;
        if (ixj > i) {
          bool up = ((i & k) == 0);
          float va = s[i], vb = s[ixj];
          if ((va > vb) == up) { s[i] = vb; s[ixj] = va; }
        }
      }
      __syncthreads();
    }
  }

  int K = Kptr[0];
  if (K < 0) K = 0;
  if (K > M_DIM - 1) K = M_DIM - 1;
  const float T = s[M_DIM - 1 - K];  // == (K+1)-th largest

  for (int j = tid; j < M_DIM; j += 256) {
    float g = src[j];
    G[(size_t)row * M_DIM + j] = (fabsf(g) > T) ? g : 0.0f;
  }
}

// ---------------------------------------------------------------------------
// Tiled transpose: dst[j][i] = src[i][j], src rows x cols
// ---------------------------------------------------------------------------
__global__ void transpose_kernel(const float* __restrict__ src, float* __restrict__ dst,
                                 int rows, int cols) {
  __shared__ float tile[32][33];
  int bx = blockIdx.x * 32, by = blockIdx.y * 32;
  int x = bx + threadIdx.x;
  for (int i = 0; i < 32; i += 8) {
    int y = by + threadIdx.y + i;
    tile[threadIdx.y + i][threadIdx.x] = src[(size_t)y * cols + x];
  }
  __syncthreads();
  x = by + threadIdx.x;
  for (int i = 0; i < 32; i += 8) {
    int y = bx + threadIdx.y + i;
    dst[(size_t)y * rows + x] = tile[threadIdx.x][threadIdx.y + i];
  }
}

// ---------------------------------------------------------------------------
// Host-side launch sequence (graph-capture safe, all on `stream`)
// ---------------------------------------------------------------------------
extern "C" void kernel_launch(void* const* d_in, const int* in_sizes, int n_in,
                              void* d_out, int out_size, void* d_ws, size_t ws_size,
                              hipStream_t stream) {
  (void)in_sizes; (void)n_in; (void)out_size; (void)ws_size;

  const float* dY = (const float*)d_in[0];  // [1024, 512]
  const float* dW = (const float*)d_in[1];  // [512, 1024]
  const int*   dK = (const int*)d_in[2];    // scalar

  float* dX     = (float*)d_out;                       // [512, 1024]
  float* dG     = dX + (size_t)N_DIM * M_DIM;          // [1024, 1024]
  float* dNorms = dG + (size_t)B_DIM * M_DIM;          // [50]

  float* ws      = (float*)d_ws;
  float* cn      = ws;                                  // 1024
  float* eta     = ws + 1024;                           // 1
  float* yn      = ws + 1025;                           // 1
  float* partial = ws + 1088;                           // 256
  float* Wn      = ws + 2048;                           // 512*1024
  float* WnT     = Wn  + (size_t)N_DIM * M_DIM;         // 512*1024 (as [1024,512])
  float* R       = WnT + (size_t)N_DIM * M_DIM;         // 1024*512
  float* Pre     = R   + (size_t)B_DIM * N_DIM;         // 1024*1024 (aliased GammaT)

  // 1) Normalize dictionary columns, build Wn and WnT
  colnorm_kernel<<<M_DIM / 256, 256, 0, stream>>>(dW, cn);
  normalize_kernel<<<(N_DIM * M_DIM) / 256, 256, 0, stream>>>(dW, cn, Wn, WnT);

  // 2) Power method -> eta = 1/c
  power_method_kernel<<<1, 1024, 0, stream>>>(Wn, eta);

  // 3) Yn = ||Y||
  sq_partial_kernel<<<256, 256, 0, stream>>>(dY, B_DIM * N_DIM, partial);
  sq_finalize_kernel<<<1, 256, 0, stream>>>(partial, 256, yn, yn, 0);

  const dim3 blk(128);
  // NT=4 kernels: block covers 64 x 128 of C
  const dim3 gridMM(M_DIM / 128, B_DIM / 64);  // C [1024,1024]: 8x16 blocks = 512 waves
  // NT=2 kernels: block covers 64 x 64 of C
  const dim3 gridRes(N_DIM / 64, B_DIM / 64);  // C [1024, 512]: 8x16 blocks = 512 waves
  const dim3 gridX(M_DIM / 64, N_DIM / 64);    // C [ 512,1024]: 16x8 blocks = 512 waves

  // 4) Gamma0 = HT( Y @ (eta * Wn) )   : A=Y [1024,512], Bt=WnT [1024,512]
  gemm_wmma_kernel<4><<<gridMM, blk, 0, stream>>>(Pre, M_DIM, dY, N_DIM, WnT, N_DIM,
                                                  dG, dY, eta, N_DIM, 0);
  hard_threshold_kernel<<<B_DIM, 256, 0, stream>>>(Pre, dG, dK);

  // 5) residual = Gamma @ Wn^T - Y     : A=Gamma [1024,1024], Bt=Wn [512,1024]
  gemm_wmma_kernel<2><<<gridRes, blk, 0, stream>>>(R, N_DIM, dG, M_DIM, Wn, M_DIM,
                                                   dG, dY, eta, M_DIM, 2);

  // 6) IHT loop
  for (int it = 0; it < IHT_STEPS; ++it) {
    // Pre = Gamma - eta * (R @ Wn)     : A=R [1024,512], Bt=WnT
    gemm_wmma_kernel<4><<<gridMM, blk, 0, stream>>>(Pre, M_DIM, R, N_DIM, WnT, N_DIM,
                                                    dG, dY, eta, N_DIM, 1);
    hard_threshold_kernel<<<B_DIM, 256, 0, stream>>>(Pre, dG, dK);
    // residual
    gemm_wmma_kernel<2><<<gridRes, blk, 0, stream>>>(R, N_DIM, dG, M_DIM, Wn, M_DIM,
                                                     dG, dY, eta, M_DIM, 2);
    // norms[it] = ||residual|| / Yn
    sq_partial_kernel<<<256, 256, 0, stream>>>(R, B_DIM * N_DIM, partial);
    sq_finalize_kernel<<<1, 256, 0, stream>>>(partial, 256, dNorms + it, yn, 1);
  }

  // 7) X = Wn @ Gamma : A=Wn [512,1024], Bt=Gamma^T (transpose into Pre)
  transpose_kernel<<<dim3(M_DIM / 32, B_DIM / 32), dim3(32, 8), 0, stream>>>(dG, Pre,
                                                                             B_DIM, M_DIM);
  gemm_wmma_kernel<2><<<gridX, blk, 0, stream>>>(dX, M_DIM, Wn, M_DIM, Pre, M_DIM,
                                                 dG, dY, eta, M_DIM, 3);
}